// tPCN_37417755083686
// MI455X (gfx1250) — compile-verified
//
#include <hip/hip_runtime.h>
#include <math.h>

#define H        64
#define NDIR     4
#define NOBS     5
#define MAX_ITER 10
#define TOL      1e-8f
#define LR       1e-3f
#define EPSN     1e-6f
#define SSTRIDE  68   // padded LDS row stride in words (bank-conflict free, 16B-aligned rows)

typedef float v2f __attribute__((ext_vector_type(2)));
typedef float v8f __attribute__((ext_vector_type(8)));

__device__ __forceinline__ void lds_fence() {
    // CDNA5 split dep counter: wait for all outstanding LDS ops (wave-internal
    // cross-lane LDS transpose ordering).
    asm volatile("s_wait_dscnt 0" ::: "memory");
}

__global__ __launch_bounds__(128)
void tpcn_kernel(const float* __restrict__ prev_state,
                 const float* __restrict__ Wr,
                 const float* __restrict__ Win,
                 const float* __restrict__ Wout,
                 const int*   __restrict__ dirv,
                 const int*   __restrict__ obsv,
                 float*       __restrict__ out,
                 int Btot)
{
    // Block = 4 waves; each wave owns 16 batch rows.
    __shared__ float sWr[64 * SSTRIDE];     // Wr row-major, padded
    __shared__ float sWoutT[64 * 8];        // Wout^T, per-column 5 values padded to 8
    __shared__ float sWin[64 * NDIR];       // Win row-major
    __shared__ float sPrev[4][16 * SSTRIDE];// per-wave prev_s tile (WMMA A source)
    __shared__ float sG[4][16 * SSTRIDE];   // per-wave matmul-result staging

    const int tid = threadIdx.x;

    for (int i = tid; i < 64 * 64; i += 128)
        sWr[(i >> 6) * SSTRIDE + (i & 63)] = Wr[i];
    for (int i = tid; i < NOBS * 64; i += 128) {
        int o = i >> 6, c = i & 63;
        sWoutT[c * 8 + o] = Wout[i];
    }
    for (int i = tid; i < 64 * NDIR; i += 128)
        sWin[i] = Win[i];
    __syncthreads();

    const int wave = tid >> 5;
    const int lane = tid & 31;
    const int row  = lane & 15;   // batch row within the wave tile
    const int hi   = lane >> 4;   // column-half owned by this lane (cols 32*hi..+31)
    const int r0   = (blockIdx.x * 4 + wave) * 16;
    const int myrow = r0 + row;
    const int rclamp = (myrow < Btot) ? myrow : (Btot - 1);

    float* SP = &sPrev[wave][0];
    float* SGm = &sG[wave][0];

    const int dir_own = dirv[rclamp];
    const int obs_own = obsv[rclamp];

    // ---- stage prev_state tile to LDS; track row-sum sp
    float sp = 0.f;
    {
        const float* src = prev_state + (size_t)rclamp * H + 32 * hi;
        #pragma unroll
        for (int k = 0; k < 8; ++k) {
            float4 x = *(const float4*)(src + 4 * k);
            *(float4*)&SP[row * SSTRIDE + 32 * hi + 4 * k] = x;
            sp += x.x + x.y + x.z + x.w;
        }
    }
    sp += __shfl_xor(sp, 16);

    // ---- dir embedding directly in WMMA C/D layout (accumulator init)
    v8f de8[4];
    #pragma unroll
    for (int t = 0; t < 4; ++t) {
        #pragma unroll
        for (int j = 0; j < 8; ++j) {
            int rr = j + 8 * hi;                 // matrix row this (vgpr,lane-half) holds
            int d  = __shfl(dir_own, rr);        // dir of that batch row
            de8[t][j] = sWin[(16 * t + row) * NDIR + d];
        }
    }

    float cur[32];
    float g[32];
    float sc;

    // g = relu(SP(16x64) @ Wr^T + dir_embed), delivered as this lane's row slice.
    auto matmul_g = [&]() {
        v8f acc[4];
        #pragma unroll
        for (int t = 0; t < 4; ++t) acc[t] = de8[t];
        lds_fence();   // SP writes must be visible to cross-lane A-fragment reads
        #pragma unroll
        for (int kb = 0; kb < 16; ++kb) {
            // A fragment (16x4 f32 layout): K = 4kb + 2*hi + {0,1}
            v2f a = *(const v2f*)&SP[row * SSTRIDE + kb * 4 + 2 * hi];
            #pragma unroll
            for (int t = 0; t < 4; ++t) {
                // B fragment (4x16 K x N): B[k][n] = Wr[n][k], n = 16t + row
                v2f b = *(const v2f*)&sWr[(16 * t + row) * SSTRIDE + kb * 4 + 2 * hi];
                acc[t] = __builtin_amdgcn_wmma_f32_16x16x4_f32(
                    false, a, false, b, (short)0, acc[t], false, false);
            }
        }
        // relu + scatter C/D layout -> row-major staging
        #pragma unroll
        for (int t = 0; t < 4; ++t) {
            #pragma unroll
            for (int j = 0; j < 8; ++j)
                SGm[(j + 8 * hi) * SSTRIDE + 16 * t + row] = fmaxf(acc[t][j], 0.f);
        }
        lds_fence();
        #pragma unroll
        for (int k = 0; k < 8; ++k) {
            float4 x = *(const float4*)&SGm[row * SSTRIDE + 32 * hi + 4 * k];
            g[4 * k + 0] = x.x; g[4 * k + 1] = x.y;
            g[4 * k + 2] = x.z; g[4 * k + 3] = x.w;
        }
    };

    float fr[NOBS];
    auto softmax_cur = [&]() {
        float l[NOBS] = {0.f, 0.f, 0.f, 0.f, 0.f};
        #pragma unroll
        for (int i = 0; i < 32; ++i) {
            int c = 32 * hi + i;
            float4 w4 = *(const float4*)&sWoutT[c * 8];
            float  w5 = sWoutT[c * 8 + 4];
            l[0] = fmaf(cur[i], w4.x, l[0]);
            l[1] = fmaf(cur[i], w4.y, l[1]);
            l[2] = fmaf(cur[i], w4.z, l[2]);
            l[3] = fmaf(cur[i], w4.w, l[3]);
            l[4] = fmaf(cur[i], w5,   l[4]);
        }
        #pragma unroll
        for (int o = 0; o < NOBS; ++o) l[o] += __shfl_xor(l[o], 16);
        float mx = l[0];
        #pragma unroll
        for (int o = 1; o < NOBS; ++o) mx = fmaxf(mx, l[o]);
        float s = 0.f;
        #pragma unroll
        for (int o = 0; o < NOBS; ++o) { fr[o] = __expf(l[o] - mx); s += fr[o]; }
        float is = 1.f / s;
        #pragma unroll
        for (int o = 0; o < NOBS; ++o) fr[o] *= is;
    };

    // ---- init: cur = normalize(relu(prev_state @ Wr^T + de))
    matmul_g();
    {
        float ss = 0.f, sv = 0.f;
        #pragma unroll
        for (int i = 0; i < 32; ++i) { ss = fmaf(g[i], g[i], ss); sv += g[i]; }
        ss += __shfl_xor(ss, 16);
        sv += __shfl_xor(sv, 16);
        float inv = 1.f / (sqrtf(ss) + EPSN);
        #pragma unroll
        for (int i = 0; i < 32; ++i) cur[i] = g[i] * inv;
        sc = sv * inv;
    }

    // ---- relaxation iterations
    bool active = true;
    for (int it = 0; it < MAX_ITER; ++it) {
        bool cond = active && (fabsf(sp - sc) * (1.f / 64.f) > TOL);
        if (!__any((int)cond)) break;      // wave-uniform early exit (EXEC stays full)

        matmul_g();                        // g = relu(prev_s @ Wr^T + de)
        softmax_cur();                     // fr = softmax(cur @ Wout^T)

        float dot = 0.f, ep[NOBS], jv[NOBS];
        #pragma unroll
        for (int o = 0; o < NOBS; ++o) {
            ep[o] = ((o == obs_own) ? 1.f : 0.f) - fr[o];
            dot = fmaf(fr[o], ep[o], dot);
        }
        #pragma unroll
        for (int o = 0; o < NOBS; ++o) jv[o] = fr[o] * ep[o] - fr[o] * dot;

        float ss = 0.f, sv = 0.f;
        #pragma unroll
        for (int i = 0; i < 32; ++i) {
            int c = 32 * hi + i;
            float4 w4 = *(const float4*)&sWoutT[c * 8];
            float  w5 = sWoutT[c * 8 + 4];
            float dd = -0.1f * (cur[i] - g[i]);       // -0.1 * eps_g
            dd = fmaf(jv[0], w4.x, dd);
            dd = fmaf(jv[1], w4.y, dd);
            dd = fmaf(jv[2], w4.z, dd);
            dd = fmaf(jv[3], w4.w, dd);
            dd = fmaf(jv[4], w5,   dd);
            float v = fmaf(LR, dd, cur[i]);
            g[i] = v;                                  // reuse g[] as new_cur (pre-norm)
            ss = fmaf(v, v, ss);
            sv += v;
        }
        ss += __shfl_xor(ss, 16);
        sv += __shfl_xor(sv, 16);
        float inv = 1.f / (sqrtf(ss) + EPSN);

        if (cond) {
            // prev_s <- cur (old): refresh this row's slice in the A-source tile
            #pragma unroll
            for (int k = 0; k < 8; ++k) {
                float4 x = make_float4(cur[4 * k], cur[4 * k + 1],
                                       cur[4 * k + 2], cur[4 * k + 3]);
                *(float4*)&SP[row * SSTRIDE + 32 * hi + 4 * k] = x;
            }
            sp = sc;
            #pragma unroll
            for (int i = 0; i < 32; ++i) cur[i] = g[i] * inv;
            sc = sv * inv;
        }
        active = cond;
    }

    // ---- outputs: next_obs_pred = softmax(cur @ Wout^T), then cur
    softmax_cur();
    if (myrow < Btot) {
        if (hi == 0) {
            float* po = out + (size_t)myrow * NOBS;
            #pragma unroll
            for (int o = 0; o < NOBS; ++o) po[o] = fr[o];
        }
        float* co = out + (size_t)Btot * NOBS + (size_t)myrow * H + 32 * hi;
        #pragma unroll
        for (int k = 0; k < 8; ++k)
            *(float4*)(co + 4 * k) = make_float4(cur[4 * k], cur[4 * k + 1],
                                                 cur[4 * k + 2], cur[4 * k + 3]);
    }
}

extern "C" void kernel_launch(void* const* d_in, const int* in_sizes, int n_in,
                              void* d_out, int out_size, void* d_ws, size_t ws_size,
                              hipStream_t stream) {
    const float* prev_state = (const float*)d_in[0];
    const float* Wr         = (const float*)d_in[1];
    const float* Win        = (const float*)d_in[2];
    const float* Wout       = (const float*)d_in[3];
    const int*   dirv       = (const int*)d_in[4];
    const int*   obsv       = (const int*)d_in[5];
    float*       out        = (float*)d_out;

    int Btot = in_sizes[4];                 // direction has B elements
    int rowsPerBlock = 64;                  // 4 waves x 16 rows
    int grid = (Btot + rowsPerBlock - 1) / rowsPerBlock;

    tpcn_kernel<<<grid, 128, 0, stream>>>(prev_state, Wr, Win, Wout,
                                          dirv, obsv, out, Btot);
}